// projection_NN_85469849190924
// MI455X (gfx1250) — compile-verified
//
#include <hip/hip_runtime.h>
#include <hip/hip_bf16.h>

// ---------------------------------------------------------------------------
// Problem constants (match reference)
// ---------------------------------------------------------------------------
#define NDIM   32
#define MDIM   8
#define TAU    50
#define HID    512
#define NCON   800
#define BATCH  1024
#define DDIM   (MDIM * TAU)   // 400
#define ADMM_ITERS 40
#define NS_ITERS   25
#define RHO_F  1.0f

typedef float f32x2 __attribute__((ext_vector_type(2)));
typedef float f32x8 __attribute__((ext_vector_type(8)));

enum {
  EPI_NONE = 0,
  EPI_BIAS_RELU,        // relu(acc + aux0[col])
  EPI_BIAS_TANH_SCALE,  // tanh(acc + aux0[col]) * aux1[col & 7]
  EPI_ROWVEC_ADD,       // acc + aux0[col]
  EPI_RHO_ADD_2I,       // alpha*acc + 2*(row==col)
  EPI_2I_MINUS,         // 2*(row==col) - acc
  EPI_2V_PLUS_RHO,      // 2*aux0[row*ldc+col] + alpha*acc
  EPI_MIN_AUX           // min(acc, aux0[row*ldc+col])
};

// Fragment load along K: CONT -> contiguous pair (global_load_b64),
// else two strided b32 loads (coalesced across lanes).
template <bool CONT>
__device__ __forceinline__ f32x2 ldfrag(const float* __restrict__ p, long ka, long es) {
  if (CONT) {
    return *reinterpret_cast<const f32x2*>(p + ka);   // ka even, es==1, base even -> 8B aligned
  } else {
    f32x2 r;
    r.x = p[ka * es];
    r.y = p[(ka + 1) * es];
    return r;
  }
}

template <int EPI>
__device__ __forceinline__ void store_tile(float* __restrict__ C, int ldc, int mbase, int col,
                                           int half, f32x8 acc,
                                           const float* __restrict__ aux0,
                                           const float* __restrict__ aux1, float alpha) {
#pragma unroll
  for (int v = 0; v < 8; ++v) {
    const int row = mbase + v + half * 8;
    float x = acc[v];
    if (EPI == EPI_BIAS_RELU)       x = fmaxf(x + aux0[col], 0.0f);
    if (EPI == EPI_BIAS_TANH_SCALE) x = tanhf(x + aux0[col]) * aux1[col & 7];
    if (EPI == EPI_ROWVEC_ADD)      x = x + aux0[col];
    if (EPI == EPI_RHO_ADD_2I)      x = alpha * x + ((row == col) ? 2.0f : 0.0f);
    if (EPI == EPI_2I_MINUS)        x = ((row == col) ? 2.0f : 0.0f) - x;
    if (EPI == EPI_2V_PLUS_RHO)     x = 2.0f * aux0[(long)row * ldc + col] + alpha * x;
    if (EPI == EPI_MIN_AUX)         x = fminf(x, aux0[(long)row * ldc + col]);
    C[(long)row * ldc + col] = x;
  }
}

// ---------------------------------------------------------------------------
// fp32 WMMA GEMM: C[M,N] (+epi) = A[M,K] * B[K,N]
// A addressed as A[i*a_rs + k*a_cs], B as B[k*b_rs + j*b_cs]  -> free transposes.
// One wave32 per 32x32 macro-tile (4 accumulators, 2 A-frags + 2 B-frags per
// K-step feed 4 WMMAs -> 1 fragment load per WMMA). Edge macro-tiles clamp
// the second sub-tile (wave-uniform) and guard stores, so EXEC is all-ones at
// every V_WMMA (ISA 7.12 requirement).
// Fragment layouts per CDNA5 ISA 7.12.2:
//   A 16x4 : lane m = lane&15, VGPR v holds K = 2*(lane>=16)+v
//   B 4x16 : lane n = lane&15, VGPR v holds K = 2*(lane>=16)+v
//   C/D    : lane n = lane&15, VGPR v holds M = 8*(lane>=16)+v
// ---------------------------------------------------------------------------
template <int EPI, bool ACONT, bool BCONT>
__global__ __launch_bounds__(256) void gemm_wmma_f32(
    const float* __restrict__ A, int a_rs, int a_cs,
    const float* __restrict__ B, int b_rs, int b_cs,
    float* __restrict__ C, int ldc,
    int Mdim, int Ndim, int Kdim,
    const float* __restrict__ aux0, const float* __restrict__ aux1,
    float alpha)
{
  const int Mt2 = (Mdim + 31) >> 5;
  const int Nt2 = (Ndim + 31) >> 5;
  const int wid = (blockIdx.x * blockDim.x + threadIdx.x) >> 5;
  if (wid >= Mt2 * Nt2) return;               // wave-uniform exit
  const int m0 = (wid / Nt2) << 5;
  const int n0 = (wid % Nt2) << 5;
  const bool hasM1 = (m0 + 16 < Mdim);
  const bool hasN1 = (n0 + 16 < Ndim);
  const int m1 = hasM1 ? m0 + 16 : m0;        // clamp -> duplicate compute on edges
  const int n1 = hasN1 ? n0 + 16 : n0;

  const int lane = threadIdx.x & 31;
  const int half = lane >> 4;                 // 0: lanes 0-15, 1: lanes 16-31
  const int lm   = lane & 15;
  const int kh   = half * 2;

  const float* __restrict__ A0 = A + (long)(m0 + lm) * a_rs;
  const float* __restrict__ A1 = A + (long)(m1 + lm) * a_rs;
  const float* __restrict__ B0 = B + (long)(n0 + lm) * b_cs;
  const float* __restrict__ B1 = B + (long)(n1 + lm) * b_cs;

  f32x8 acc00 = {0.f,0.f,0.f,0.f,0.f,0.f,0.f,0.f};
  f32x8 acc01 = acc00, acc10 = acc00, acc11 = acc00;

#pragma unroll 2
  for (int k = 0; k < Kdim; k += 4) {
    const long ka = k + kh;
    const f32x2 a0 = ldfrag<ACONT>(A0, ka, a_cs);
    const f32x2 a1 = ldfrag<ACONT>(A1, ka, a_cs);
    const f32x2 b0 = ldfrag<BCONT>(B0, ka, b_rs);
    const f32x2 b1 = ldfrag<BCONT>(B1, ka, b_rs);
    acc00 = __builtin_amdgcn_wmma_f32_16x16x4_f32(false, a0, false, b0, (short)0, acc00, false, false);
    acc01 = __builtin_amdgcn_wmma_f32_16x16x4_f32(false, a0, false, b1, (short)0, acc01, false, false);
    acc10 = __builtin_amdgcn_wmma_f32_16x16x4_f32(false, a1, false, b0, (short)0, acc10, false, false);
    acc11 = __builtin_amdgcn_wmma_f32_16x16x4_f32(false, a1, false, b1, (short)0, acc11, false, false);
  }

  store_tile<EPI>(C, ldc, m0, n0 + lm, half, acc00, aux0, aux1, alpha);
  if (hasN1)          store_tile<EPI>(C, ldc, m0, n1 + lm, half, acc01, aux0, aux1, alpha);
  if (hasM1)          store_tile<EPI>(C, ldc, m1, n0 + lm, half, acc10, aux0, aux1, alpha);
  if (hasM1 && hasN1) store_tile<EPI>(C, ldc, m1, n1 + lm, half, acc11, aux0, aux1, alpha);
}

// ---------------------------------------------------------------------------
// Helper kernels
// ---------------------------------------------------------------------------

// max_r sum_j |K[r,j]|  -> *s  (positive floats: int atomicMax is order-preserving)
__global__ void rowsum_max_kernel(const float* __restrict__ Km, int n, float* __restrict__ s) {
  int r = blockIdx.x * blockDim.x + threadIdx.x;
  if (r < n) {
    float acc = 0.0f;
    for (int j = 0; j < n; ++j) acc += fabsf(Km[(long)r * n + j]);
    atomicMax((int*)s, __float_as_int(acc));
  }
}

// X = (1/s) * I
__global__ void init_diag_kernel(float* __restrict__ X, const float* __restrict__ s, int n) {
  int i = blockIdx.x * blockDim.x + threadIdx.x;
  if (i < n * n) {
    int r = i / n, c = i % n;
    X[i] = (r == c) ? (1.0f / s[0]) : 0.0f;
  }
}

// y = 0, w = z
__global__ void init_wy_kernel(const float* __restrict__ z, float* __restrict__ w,
                               float* __restrict__ y, int n) {
  int i = blockIdx.x * blockDim.x + threadIdx.x;
  if (i < n) { w[i] = z[i]; y[i] = 0.0f; }
}

// z' = min(Fu + y, b); y' = y + Fu - z'; w = z' - y'
__global__ void admm_update_kernel(const float* __restrict__ Fu, const float* __restrict__ bm,
                                   float* __restrict__ z, float* __restrict__ y,
                                   float* __restrict__ w, int n) {
  int i = blockIdx.x * blockDim.x + threadIdx.x;
  if (i < n) {
    float fu = Fu[i], yy = y[i];
    float zn = fminf(fu + yy, bm[i]);
    float yn = yy + fu - zn;
    z[i] = zn; y[i] = yn; w[i] = zn - yn;
  }
}

// out[c*M + r] = u[r*D + c]
__global__ void transpose_out_kernel(const float* __restrict__ u, float* __restrict__ out,
                                     int M, int D) {
  int i = blockIdx.x * blockDim.x + threadIdx.x;
  if (i < M * D) {
    int r = i / D, c = i % D;
    out[(long)c * M + r] = u[i];
  }
}

// ---------------------------------------------------------------------------
// Host launch
// ---------------------------------------------------------------------------
template <int EPI>
static void launch_gemm(const float* A, int a_rs, int a_cs,
                        const float* B, int b_rs, int b_cs,
                        float* C, int ldc, int M, int N, int K,
                        const float* aux0, const float* aux1, float alpha,
                        hipStream_t stream) {
  const int mt2 = (M + 31) / 32, nt2 = (N + 31) / 32;
  const int thr = mt2 * nt2 * 32;
  const int blk = (thr + 255) / 256;
  const bool ac = (a_cs == 1), bc = (b_rs == 1);
  if (ac && bc)
    gemm_wmma_f32<EPI, true, true><<<blk, 256, 0, stream>>>(A, a_rs, a_cs, B, b_rs, b_cs,
                                                            C, ldc, M, N, K, aux0, aux1, alpha);
  else if (ac)
    gemm_wmma_f32<EPI, true, false><<<blk, 256, 0, stream>>>(A, a_rs, a_cs, B, b_rs, b_cs,
                                                             C, ldc, M, N, K, aux0, aux1, alpha);
  else if (bc)
    gemm_wmma_f32<EPI, false, true><<<blk, 256, 0, stream>>>(A, a_rs, a_cs, B, b_rs, b_cs,
                                                             C, ldc, M, N, K, aux0, aux1, alpha);
  else
    gemm_wmma_f32<EPI, false, false><<<blk, 256, 0, stream>>>(A, a_rs, a_cs, B, b_rs, b_cs,
                                                              C, ldc, M, N, K, aux0, aux1, alpha);
}

extern "C" void kernel_launch(void* const* d_in, const int* in_sizes, int n_in,
                              void* d_out, int out_size, void* d_ws, size_t ws_size,
                              hipStream_t stream) {
  const float* X0    = (const float*)d_in[0];   // [32, 1024] row-major
  const float* W1    = (const float*)d_in[1];   // [32, 512]
  const float* b1    = (const float*)d_in[2];   // [512]
  const float* W2    = (const float*)d_in[3];   // [512, 512]
  const float* b2    = (const float*)d_in[4];   // [512]
  const float* W3    = (const float*)d_in[5];   // [512, 400]
  const float* b3    = (const float*)d_in[6];   // [400]
  const float* u_bar = (const float*)d_in[7];   // [8, 1]
  const float* F     = (const float*)d_in[8];   // [800, 400]
  const float* g     = (const float*)d_in[9];   // [800]
  const float* H0    = (const float*)d_in[10];  // [800, 32]
  float* out = (float*)d_out;                   // [400, 1024]

  // Workspace carving (~27 MB of fp32 — all L2 resident)
  float* ws = (float*)d_ws;
  float* h1 = ws;  ws += (long)BATCH * HID;
  float* h2 = ws;  ws += (long)BATCH * HID;
  float* V  = ws;  ws += (long)BATCH * DDIM;
  float* bm = ws;  ws += (long)BATCH * NCON;
  float* Km = ws;  ws += (long)DDIM * DDIM;
  float* Xa = ws;  ws += (long)DDIM * DDIM;
  float* Xb = ws;  ws += (long)DDIM * DDIM;
  float* Tm = ws;  ws += (long)DDIM * DDIM;
  float* z  = ws;  ws += (long)BATCH * NCON;
  float* y  = ws;  ws += (long)BATCH * NCON;
  float* w  = ws;  ws += (long)BATCH * NCON;
  float* Tv = ws;  ws += (long)BATCH * DDIM;
  float* u  = ws;  ws += (long)BATCH * DDIM;
  float* Fu = ws;  ws += (long)BATCH * NCON;
  float* sn = ws;  ws += 1;

  // ---- MLP head:  h1 = relu(X0^T W1 + b1); h2 = relu(h1 W2 + b2);
  //                 V  = tanh(h2 W3 + b3) * u_bar[col % 8]
  launch_gemm<EPI_BIAS_RELU>(X0, 1, BATCH, W1, HID, 1, h1, HID, BATCH, HID, NDIM,
                             b1, nullptr, 0.f, stream);
  launch_gemm<EPI_BIAS_RELU>(h1, HID, 1, W2, HID, 1, h2, HID, BATCH, HID, HID,
                             b2, nullptr, 0.f, stream);
  launch_gemm<EPI_BIAS_TANH_SCALE>(h2, HID, 1, W3, DDIM, 1, V, DDIM, BATCH, DDIM, HID,
                                   b3, u_bar, 0.f, stream);

  // ---- Constraint RHS: bm = g + X0^T H0^T       (B[k][j] = H0[j*32+k])
  launch_gemm<EPI_ROWVEC_ADD>(X0, 1, BATCH, H0, 1, NDIM, bm, NCON, BATCH, NCON, NDIM,
                              g, nullptr, 0.f, stream);

  // ---- K = 2I + rho * F^T F   (A[i][k] = F[k*D+i])
  launch_gemm<EPI_RHO_ADD_2I>(F, 1, DDIM, F, DDIM, 1, Km, DDIM, DDIM, DDIM, NCON,
                              nullptr, nullptr, RHO_F, stream);

  // ---- Kinv via Newton–Schulz: X0 = I/||K||_inf; X <- X(2I - K X)
  hipMemsetAsync(sn, 0, sizeof(float), stream);
  rowsum_max_kernel<<<(DDIM + 255) / 256, 256, 0, stream>>>(Km, DDIM, sn);
  init_diag_kernel<<<(DDIM * DDIM + 255) / 256, 256, 0, stream>>>(Xa, sn, DDIM);
  float* Xc = Xa; float* Xn = Xb;
  for (int it = 0; it < NS_ITERS; ++it) {
    launch_gemm<EPI_2I_MINUS>(Km, DDIM, 1, Xc, DDIM, 1, Tm, DDIM, DDIM, DDIM, DDIM,
                              nullptr, nullptr, 0.f, stream);
    launch_gemm<EPI_NONE>(Xc, DDIM, 1, Tm, DDIM, 1, Xn, DDIM, DDIM, DDIM, DDIM,
                          nullptr, nullptr, 0.f, stream);
    float* t = Xc; Xc = Xn; Xn = t;
  }
  const float* Kinv = Xc;

  // ---- z = min(V F^T, bm); y = 0; w = z - y
  launch_gemm<EPI_MIN_AUX>(V, DDIM, 1, F, 1, DDIM, z, NCON, BATCH, NCON, DDIM,
                           bm, nullptr, 0.f, stream);
  init_wy_kernel<<<(BATCH * NCON + 255) / 256, 256, 0, stream>>>(z, w, y, BATCH * NCON);

  // ---- ADMM: 41 u-updates (final one included), 40 z/y updates
  for (int it = 0; it <= ADMM_ITERS; ++it) {
    // Tv = 2V + rho * (w F);  u = Tv Kinv
    launch_gemm<EPI_2V_PLUS_RHO>(w, NCON, 1, F, DDIM, 1, Tv, DDIM, BATCH, DDIM, NCON,
                                 V, nullptr, RHO_F, stream);
    launch_gemm<EPI_NONE>(Tv, DDIM, 1, Kinv, DDIM, 1, u, DDIM, BATCH, DDIM, DDIM,
                          nullptr, nullptr, 0.f, stream);
    if (it < ADMM_ITERS) {
      // Fu = u F^T; z/y/w elementwise update
      launch_gemm<EPI_NONE>(u, DDIM, 1, F, 1, DDIM, Fu, NCON, BATCH, NCON, DDIM,
                            nullptr, nullptr, 0.f, stream);
      admm_update_kernel<<<(BATCH * NCON + 255) / 256, 256, 0, stream>>>(Fu, bm, z, y, w,
                                                                         BATCH * NCON);
    }
  }

  // ---- out = u^T  [400, 1024]
  transpose_out_kernel<<<(BATCH * DDIM + 255) / 256, 256, 0, stream>>>(u, out, BATCH, DDIM);
}